// Sinosmth_grad_k_1159641170445
// MI455X (gfx1250) — compile-verified
//
#include <hip/hip_runtime.h>

// Sino smoothing: per-row tridiagonal solve (forward sweep + back-substitution)
// B = 65536 rows, N = 736. One wave32 lane per row; 32 rows per 32-thread block.
// y/beta tiles are DMA'd into LDS with the CDNA5 Tensor Data Mover
// (tensor_load_to_lds + s_wait_tensorcnt), double-buffered, TDM LDS padding
// (33-dword rows) for bank-conflict-free reads. xs/c spills and the backward
// pass use B128-vectorized global traffic (L2-resident round trip).

typedef __attribute__((ext_vector_type(4))) unsigned int v4u;
typedef __attribute__((ext_vector_type(8))) int         v8i;
typedef __attribute__((ext_vector_type(4))) int         v4i;

#define NCOLS 736
#define TILE  32
#define NT    (NCOLS / TILE)   // 23 tiles, exact
#define ROWS  32               // rows per block == blockDim.x
#define LROW  33               // padded LDS row stride (dwords)

#if __has_builtin(__builtin_amdgcn_tensor_load_to_lds)
#define USE_TDM 1
#else
#define USE_TDM 0
#endif

// Load a 32x32 float tile (row stride NCOLS elements in memory) into LDS at
// byte offset lds_off, padded to 33-dword rows.
__device__ __forceinline__ void load_tile(float* ldsbuf, unsigned lds_off,
                                          const float* gsrc, int lane) {
#if USE_TDM
  unsigned long long ga = (unsigned long long)gsrc;
  // ---- D# group 0 (ISA 8.3): count=1 valid; lds_addr; 57-bit global addr; type=2
  v4u g0;
  g0.x = 1u;                                                  // count=1, is_restore=0, gather off
  g0.y = lds_off;                                             // LDS byte address
  g0.z = (unsigned)ga;                                        // global_addr[31:0]
  g0.w = (unsigned)((ga >> 32) & 0x01FFFFFFull) | 0x80000000u;// global_addr[56:32] | type=2
  // ---- D# group 1 (ISA 8.4)
  v8i g1;
  g1[0] = (int)((2u << 16)      // data_size = 2 -> 4 bytes
              | (1u << 20)      // pad_enable
              | (4u << 22));    // pad_interval = 4 -> every 32 dwords; pad_amount=0 -> 1 dword
  g1[1] = 0;                    // atomic_barrier_addr=0; tensor_dim0[15:0] (td0 = 1<<30)
  g1[2] = 0x4000;               // tensor_dim0[31:16] = 0x4000 (huge, no OOB); td1[15:0]=0
  g1[3] = (int)(0x4000u | ((unsigned)TILE << 16)); // td1[31:16]=0x4000; tile_dim0 = 32
  g1[4] = TILE;                 // tile_dim1 = 32; tile_dim2 = 0 (2D tile)
  g1[5] = NCOLS;                // tensor_dim0_stride[31:0] = 736 elements
  g1[6] = 0;                    // stride0[47:32]=0; stride1[15:0]=0
  g1[7] = 0;                    // stride1[47:16]=0 (unused, tile_dim2==0)
  v4i gz = {0, 0, 0, 0};        // groups 2/3 unused for 2D
#if defined(__clang_major__) && (__clang_major__ >= 23)
  v8i gz8 = {0, 0, 0, 0, 0, 0, 0, 0};
  __builtin_amdgcn_tensor_load_to_lds(g0, g1, gz, gz, gz8, 0);
#else
  __builtin_amdgcn_tensor_load_to_lds(g0, g1, gz, gz, 0);
#endif
  __asm__ volatile("" ::: "memory");
  (void)ldsbuf; (void)lane;
#else
  // Fallback: cooperative coalesced staging (lane j loads column j of each row).
  #pragma unroll 8
  for (int r = 0; r < ROWS; ++r)
    ldsbuf[r * LROW + lane] = gsrc[(long long)r * NCOLS + lane];
  (void)lds_off;
#endif
}

__device__ __forceinline__ void wait_tiles(int pending) {
#if USE_TDM
  if (pending)
    __builtin_amdgcn_s_wait_tensorcnt(2);   // current tile landed; next tile in flight
  else
    __builtin_amdgcn_s_wait_tensorcnt(0);   // everything landed
  __asm__ volatile("" ::: "memory");
#else
  (void)pending;  // normal LDS writes: compiler tracks DScnt hazards
#endif
}

// One forward elimination step. Carry (k, kfy) with kfy == k*fy.
__device__ __forceinline__ void fstep(float yv, float bv, float& k, float& kfy,
                                      float& xs, float& c) {
  float num = kfy + yv;
  float den = k + 1.0f;
  float d   = den + bv;
  float rd  = 1.0f / d;          // single reciprocal per element
  xs  = num * rd;                // provisional solution
  c   = bv * rd;                 // back-substitution coefficient
  k   = den * c;
  kfy = c * num;                 // == k_new * fy_new
}

extern "C" __global__ __launch_bounds__(32)
void sino_smooth_kernel(const float* __restrict__ y, const float* __restrict__ beta,
                        float* __restrict__ out, float* __restrict__ ws) {
  __shared__ float sY[2][ROWS * LROW];   // 2 x 4224 B
  __shared__ float sB[2][ROWS * LROW];   // 2 x 4224 B  (16.5 KB total -> many blocks/WGP)

  const int lane = (int)threadIdx.x;
  const long long row0 = (long long)blockIdx.x * ROWS;
  const float* yb = y    + row0 * NCOLS;
  const float* bb = beta + row0 * NCOLS;
  float* __restrict__ orow = out + (row0 + lane) * NCOLS;  // 2944 B stride -> 16B aligned
  float* __restrict__ wrow = ws  + (row0 + lane) * NCOLS;

  unsigned oY[2], oB[2];   // LDS byte offsets (flat addr low 32 bits == LDS offset)
  oY[0] = (unsigned)(unsigned long long)&sY[0][0];
  oY[1] = (unsigned)(unsigned long long)&sY[1][0];
  oB[0] = (unsigned)(unsigned long long)&sB[0][0];
  oB[1] = (unsigned)(unsigned long long)&sB[1][0];

  // Prologue: kick off tile 0.
  load_tile(&sY[0][0], oY[0], yb, lane);
  load_tile(&sB[0][0], oB[0], bb, lane);

  float k = 0.0f, kfy = 0.0f, xlast = 0.0f;

  for (int t = 0; t < NT; ++t) {
    const int buf = t & 1;
    if (t + 1 < NT) {
      load_tile(&sY[buf ^ 1][0], oY[buf ^ 1], yb + (t + 1) * TILE, lane);
      load_tile(&sB[buf ^ 1][0], oB[buf ^ 1], bb + (t + 1) * TILE, lane);
      wait_tiles(1);
    } else {
      wait_tiles(0);
    }

    const float* ly = &sY[buf][lane * LROW];
    const float* lb = &sB[buf][lane * LROW];
    const int colBase = t * TILE;
    float4* __restrict__ ov = (float4*)(orow + colBase);   // colBase % 4 == 0
    float4* __restrict__ wv = (float4*)(wrow + colBase);

    if (t < NT - 1) {
      // Full tile: 8 x (4 steps + one b128 store each for xs and c).
      #pragma unroll
      for (int q = 0; q < TILE / 4; ++q) {
        float4 xs4, c4;
        fstep(ly[4 * q + 0], lb[4 * q + 0], k, kfy, xs4.x, c4.x);
        fstep(ly[4 * q + 1], lb[4 * q + 1], k, kfy, xs4.y, c4.y);
        fstep(ly[4 * q + 2], lb[4 * q + 2], k, kfy, xs4.z, c4.z);
        fstep(ly[4 * q + 3], lb[4 * q + 3], k, kfy, xs4.w, c4.w);
        ov[q] = xs4;
        wv[q] = c4;
      }
    } else {
      // Last tile: cols 704..734 are normal steps (28 vectorized + 3 scalar),
      // col 735 is x_last.
      #pragma unroll
      for (int q = 0; q < 7; ++q) {
        float4 xs4, c4;
        fstep(ly[4 * q + 0], lb[4 * q + 0], k, kfy, xs4.x, c4.x);
        fstep(ly[4 * q + 1], lb[4 * q + 1], k, kfy, xs4.y, c4.y);
        fstep(ly[4 * q + 2], lb[4 * q + 2], k, kfy, xs4.z, c4.z);
        fstep(ly[4 * q + 3], lb[4 * q + 3], k, kfy, xs4.w, c4.w);
        ov[q] = xs4;
        wv[q] = c4;
      }
      #pragma unroll
      for (int i = 28; i < TILE - 1; ++i) {
        float xs, c;
        fstep(ly[i], lb[i], k, kfy, xs, c);
        orow[colBase + i] = xs;
        wrow[colBase + i] = c;
      }
      xlast = (kfy + ly[TILE - 1]) / (k + 1.0f);
      orow[NCOLS - 1] = xlast;
    }
  }

  // Backward substitution: x_i = xs_i + c_i * x_{i+1}.
  // Same-wave store->load to same address is order-guaranteed; xs/c round trip
  // is block-local and L2-hot. Scalar tail (cols 734..732), then 183 b128 chunks.
  float x = xlast;
  #pragma unroll
  for (int col = NCOLS - 2; col >= 732; --col) {
    x = fmaf(wrow[col], x, orow[col]);
    orow[col] = x;
  }
  float4* __restrict__ ov = (float4*)orow;
  const float4* __restrict__ wv = (const float4*)wrow;
  #pragma unroll 2
  for (int m = 732 / 4 - 1; m >= 0; --m) {
    float4 xs4 = ov[m];
    float4 c4  = wv[m];
    float4 r;
    r.w = fmaf(c4.w, x,   xs4.w);
    r.z = fmaf(c4.z, r.w, xs4.z);
    r.y = fmaf(c4.y, r.z, xs4.y);
    r.x = fmaf(c4.x, r.y, xs4.x);
    x = r.x;
    ov[m] = r;
  }
}

extern "C" void kernel_launch(void* const* d_in, const int* in_sizes, int n_in,
                              void* d_out, int out_size, void* d_ws, size_t ws_size,
                              hipStream_t stream) {
  const float* y    = (const float*)d_in[0];
  const float* beta = (const float*)d_in[1];  // [B, N, 1] contiguous == [B, N]
  float* out = (float*)d_out;
  float* ws  = (float*)d_ws;                  // needs B*N*4 bytes (~193 MB) for c
  (void)n_in; (void)out_size; (void)ws_size;

  const int Btot    = in_sizes[0] / NCOLS;    // 65536
  const int nblocks = Btot / ROWS;            // 2048 wave32 blocks
  sino_smooth_kernel<<<nblocks, dim3(ROWS), 0, stream>>>(y, beta, out, ws);
}